// SinusoidalPositionalEncoding_32057635898069
// MI455X (gfx1250) — compile-verified
//
#include <hip/hip_runtime.h>

typedef __attribute__((ext_vector_type(4))) float fvec4;

namespace {
constexpr int   D_MODEL       = 512;
constexpr int   VEC_PER_ROW   = D_MODEL / 4;             // 128 float4 chunks per row
constexpr float LOG2_BASEFREQ = -13.287712379549449f;    // log2(1e-4)
}

// Each block: 256 threads (8 wave32) = 2 rows x 128 float4-chunks.
// Thread t handles row = 2*blockIdx.x + (t>>7), chunk = t&127,
// i.e. channels [4*chunk, 4*chunk+3] = freq pairs (2*chunk, 2*chunk+1).
__global__ __launch_bounds__(256)
void pe_add_kernel(const float* __restrict__ x,
                   const float* __restrict__ pos,
                   float*       __restrict__ out,
                   int nrows)
{
    const int t     = threadIdx.x;
    const int chunk = t & (VEC_PER_ROW - 1);   // 0..127
    const int row   = blockIdx.x * 2 + (t >> 7);
    if (row >= nrows) return;

    // lat/lon for this row: 8-byte load, broadcast-coalesced across the half-block.
    const float2 ll  = ((const float2*)pos)[row];
    const float  lat = ll.x;
    const float  lon = ll.y;

    // freq_p = 2^(log2(1e-4) * p / 256) for p = 2*chunk, 2*chunk+1  -> v_exp_f32 x2
    const float fscale = LOG2_BASEFREQ * (1.0f / 256.0f);
    const float p0     = (float)(2 * chunk);
    const float f0     = __builtin_amdgcn_exp2f(fscale * p0);
    const float f1     = __builtin_amdgcn_exp2f(fscale * (p0 + 1.0f));

    // pe = [sin(lat*f0), cos(lon*f0), sin(lat*f1), cos(lon*f1)]  -> v_sin/v_cos_f32
    fvec4 pe;
    pe.x = __sinf(lat * f0);
    pe.y = __cosf(lon * f0);
    pe.z = __sinf(lat * f1);
    pe.w = __cosf(lon * f1);

    const int idx = row * VEC_PER_ROW + chunk;  // float4 index (max ~16.7M, fits int)

    // Software prefetch the x-stream a few rows ahead (gfx1250 global_prefetch_b8).
    if (row + 16 < nrows) {
        __builtin_prefetch(x + (size_t)(row + 16) * D_MODEL + 4 * chunk, 0, 1);
    }

    // Streaming 512MB with zero reuse: non-temporal B128 load + store (TH=NT)
    // so we don't thrash the 192MB L2.
    const fvec4 xv = __builtin_nontemporal_load(((const fvec4*)x) + idx);

    const fvec4 o = xv + pe;

    __builtin_nontemporal_store(o, ((fvec4*)out) + idx);
}

extern "C" void kernel_launch(void* const* d_in, const int* in_sizes, int n_in,
                              void* d_out, int out_size, void* d_ws, size_t ws_size,
                              hipStream_t stream)
{
    const float* x   = (const float*)d_in[0];   // (64, 2048, 512) fp32
    const float* pos = (const float*)d_in[1];   // (64, 2048, 2)   fp32
    float*       out = (float*)d_out;           // (64, 2048, 512) fp32

    const int nrows  = in_sizes[1] / 2;         // 64*2048 = 131072 rows
    const int blocks = (nrows + 1) / 2;         // 2 rows per 256-thread block

    pe_add_kernel<<<blocks, 256, 0, stream>>>(x, pos, out, nrows);
}